// Controller_73306501808240
// MI455X (gfx1250) — compile-verified
//
#include <hip/hip_runtime.h>
#include <math.h>

typedef __attribute__((ext_vector_type(2))) float v2f;
typedef __attribute__((ext_vector_type(4))) float v4f;
typedef __attribute__((ext_vector_type(8))) float v8f;

#define X_DIM 4096
#define H_DIM 1024
#define L_DIM 8
#define Z_DIM (X_DIM + 2 * H_DIM)   // 6144
#define G_DIM (4 * H_DIM)           // 4096 gate rows per layer
#define VT_DIM 512
#define ET_DIM 1024
#define F_DIM (L_DIM * H_DIM)       // 8192

// D = A(16x4 f32) x B(4x16 f32) + C accumulate, with the activation vector
// broadcast across all 16 B columns so every D column carries the 16 dots.
//
// K permutation trick for b128 loads: per K=8 step, lanes 0-15 carry logical
// k0..k0+3 and lanes 16-31 carry k0+4..k0+7 (contiguous 16B per lane).
// wmma #1 uses (w.x,w.y), wmma #2 uses (w.z,w.w); B mirrors the same mapping,
// so each wmma sums a permuted-but-consistent set of logical k — result exact.
__device__ __forceinline__ v8f gemv_step8(const float* __restrict__ wrow,
                                          const float* __restrict__ zvec,
                                          int k0, int off4, v8f acc)
{
    v4f w  = *(const v4f*)(wrow + k0 + off4);
    v4f zz = *(const v4f*)(zvec + k0 + off4);
    v2f a0; a0.x = w.x;  a0.y = w.y;
    v2f b0; b0.x = zz.x; b0.y = zz.y;
    acc = __builtin_amdgcn_wmma_f32_16x16x4_f32(
        false, a0, false, b0, (short)0, acc, false, false);
    v2f a1; a1.x = w.z;  a1.y = w.w;
    v2f b1; b1.x = zz.z; b1.y = zz.w;
    acc = __builtin_amdgcn_wmma_f32_16x16x4_f32(
        false, a1, false, b1, (short)0, acc, false, false);
    return acc;
}

// ---------------------------------------------------------------------------
// One LSTM layer, fully fused: gates GEMV (WMMA f32 16x16x4) + activations.
// Block: 1024 threads = 32 waves; block owns a 16-row h-tile.
//   wave w: gate g = w&3, K-chunk kq = w>>2 (8 chunks of 768 along Z=6144).
// ---------------------------------------------------------------------------
__global__ __launch_bounds__(1024) void lstm_layer_kernel(
    const float* __restrict__ input_x,    // (X)
    const float* __restrict__ h_prev,     // (L,H)
    const float* __restrict__ Wg,         // (L,4H,Z)
    const float* __restrict__ bg,         // (L,4H)
    const float* __restrict__ old_state,  // (L,H)
    float* __restrict__ flat,             // (L,H) hidden outputs (workspace)
    int layer)
{
    __shared__ __align__(16) float zlds[Z_DIM];   // 24 KB
    __shared__ float part[32][16];                //  2 KB wave partials

    const int tid = threadIdx.x;

    // Stage z = [input_x ; h_prev[layer] ; hidden(layer-1)] into LDS.
    for (int i = tid; i < X_DIM; i += 1024) zlds[i] = input_x[i];
    for (int i = tid; i < H_DIM; i += 1024) {
        zlds[X_DIM + i]         = h_prev[layer * H_DIM + i];
        zlds[X_DIM + H_DIM + i] = (layer == 0) ? 0.0f : flat[(layer - 1) * H_DIM + i];
    }
    __syncthreads();

    const int wave  = tid >> 5;            // 0..31
    const int lane  = tid & 31;
    const int g     = wave & 3;            // gate: 0=i 1=f 2=o 3=s
    const int kq    = wave >> 2;           // 0..7 K-chunk
    const int hbase = blockIdx.x * 16;     // 16 h rows per block
    const int row   = g * H_DIM + hbase + (lane & 15);
    const int off4  = (lane >> 4) << 2;    // 0 for lanes 0-15, 4 for 16-31

    const float* __restrict__ wrow =
        Wg + ((size_t)layer * G_DIM + row) * Z_DIM;

    const int k_begin = kq * (Z_DIM / 8);          // chunks of 768
    const int k_end   = k_begin + (Z_DIM / 8);

    v8f acc = {};
    #pragma unroll 4
    for (int k0 = k_begin; k0 < k_end; k0 += 8)
        acc = gemv_step8(wrow, zlds, k0, off4, acc);

    // All D columns identical: lane 0 holds rows m0..m0+7, lane 16 rows +8..15.
    if ((lane & 15) == 0) {
        const int mb = (lane >> 4) * 8;
        #pragma unroll
        for (int j = 0; j < 8; ++j) part[wave][mb + j] = acc[j];
    }
    __syncthreads();

    // 16 threads finish: reduce 8 K-chunks per gate, apply LSTM gating.
    if (tid < 16) {
        float gv[4];
        #pragma unroll
        for (int gg = 0; gg < 4; ++gg) {
            float s = bg[layer * G_DIM + gg * H_DIM + hbase + tid];
            #pragma unroll
            for (int q = 0; q < 8; ++q) s += part[q * 4 + gg][tid];
            gv[gg] = s;
        }
        const float c_old = old_state[layer * H_DIM + hbase + tid];
        const float i_s = 1.0f / (1.0f + expf(-gv[0]));
        const float f_s = 1.0f / (1.0f + expf(-gv[1]));
        const float o_s = 1.0f / (1.0f + expf(-gv[2]));
        const float ns  = f_s * c_old + i_s * tanhf(gv[3]);
        flat[layer * H_DIM + hbase + tid] = o_s * tanhf(ns);
    }
}

// ---------------------------------------------------------------------------
// Head: out[0:512] = Wy@flat + by ; out[512:1536] = We@flat + be.
// 1536 rows x 8192, same WMMA-GEMV pattern. Block: 512 threads = 16 waves.
//   wave w: row-tile tl = w&3 (4 tiles/block), K-chunk kq = w>>2 (4 x 2048).
// ---------------------------------------------------------------------------
__global__ __launch_bounds__(512) void head_kernel(
    const float* __restrict__ flat,
    const float* __restrict__ Wy, const float* __restrict__ by,
    const float* __restrict__ We, const float* __restrict__ be,
    float* __restrict__ out)
{
    __shared__ __align__(16) float flds[F_DIM];   // 32 KB
    __shared__ float part[16][16];

    const int tid = threadIdx.x;
    for (int i = tid; i < F_DIM; i += 512) flds[i] = flat[i];
    __syncthreads();

    const int wave = tid >> 5;
    const int lane = tid & 31;
    const int tl   = wave & 3;
    const int kq   = wave >> 2;                 // 0..3, chunks of 2048
    const int tile = blockIdx.x * 4 + tl;       // 0..95
    const int row  = tile * 16 + (lane & 15);   // 0..1535 (512 is tile-aligned)
    const int off4 = (lane >> 4) << 2;

    const float* __restrict__ wrow =
        (row < VT_DIM) ? (Wy + (size_t)row * F_DIM)
                       : (We + (size_t)(row - VT_DIM) * F_DIM);

    const int k_begin = kq * (F_DIM / 4);
    const int k_end   = k_begin + (F_DIM / 4);

    v8f acc = {};
    #pragma unroll 4
    for (int k0 = k_begin; k0 < k_end; k0 += 8)
        acc = gemv_step8(wrow, flds, k0, off4, acc);

    if ((lane & 15) == 0) {
        const int mb = (lane >> 4) * 8;
        #pragma unroll
        for (int j = 0; j < 8; ++j) part[wave][mb + j] = acc[j];
    }
    __syncthreads();

    if (tid < 64) {
        const int tl2 = tid >> 4;
        const int m   = tid & 15;
        float s = 0.0f;
        #pragma unroll
        for (int q = 0; q < 4; ++q) s += part[q * 4 + tl2][m];
        const int row2 = (blockIdx.x * 4 + tl2) * 16 + m;
        s += (row2 < VT_DIM) ? by[row2] : be[row2 - VT_DIM];
        out[row2] = s;
    }
}

extern "C" void kernel_launch(void* const* d_in, const int* in_sizes, int n_in,
                              void* d_out, int out_size, void* d_ws, size_t ws_size,
                              hipStream_t stream) {
    (void)in_sizes; (void)n_in; (void)out_size; (void)ws_size;

    const float* input_x   = (const float*)d_in[0];
    const float* h_prev    = (const float*)d_in[1];
    const float* Wg        = (const float*)d_in[2];
    const float* bg        = (const float*)d_in[3];
    const float* old_state = (const float*)d_in[4];
    const float* Wy        = (const float*)d_in[5];
    const float* by        = (const float*)d_in[6];
    const float* We        = (const float*)d_in[7];
    const float* be        = (const float*)d_in[8];
    float* out  = (float*)d_out;
    float* flat = (float*)d_ws;   // L*H floats of hidden state

    // Layers are sequentially dependent: stream order provides the sync.
    for (int l = 0; l < L_DIM; ++l) {
        lstm_layer_kernel<<<dim3(H_DIM / 16), dim3(1024), 0, stream>>>(
            input_x, h_prev, Wg, bg, old_state, flat, l);
    }
    head_kernel<<<dim3(24), dim3(512), 0, stream>>>(flat, Wy, by, We, be, out);
}